// TrueProbabiltyFlow_8933531976199
// MI455X (gfx1250) — compile-verified
//
#include <hip/hip_runtime.h>
#include <stdint.h>

typedef float v4f __attribute__((ext_vector_type(4)));
typedef int   v4i __attribute__((ext_vector_type(4)));

// OU hyperparameters (compile-time, from the reference)
#define OU_M     1.0f
#define OU_SIGMA 0.5f
#define OU_K     1.0f
#define OU_D     512

// ---- pointer cast helpers for the async builtin ----
// diag says param0 is a generic int4*; param1 is the LDS side (AS3)
#define GEN_V4I(p) ((v4i*)(uintptr_t)(p))
#define AS3_V4I(p) ((__attribute__((address_space(3))) v4i*)(uint32_t)(uintptr_t)(p))

// ---- CDNA5 async global->LDS copy availability ----
#if defined(__has_builtin)
#  if __has_builtin(__builtin_amdgcn_global_load_async_to_lds_b128)
#    define HAS_ASYNC_LDS 1
#  else
#    define HAS_ASYNC_LDS 0
#  endif
#else
#  define HAS_ASYNC_LDS 0
#endif

__device__ __forceinline__ void wait_asynccnt_le1() {
#if defined(__has_builtin) && __has_builtin(__builtin_amdgcn_s_wait_asynccnt)
    __builtin_amdgcn_s_wait_asynccnt(1);
#else
    asm volatile("s_wait_asynccnt 1" ::: "memory");
#endif
}
__device__ __forceinline__ void wait_asynccnt_le0() {
#if defined(__has_builtin) && __has_builtin(__builtin_amdgcn_s_wait_asynccnt)
    __builtin_amdgcn_s_wait_asynccnt(0);
#else
    asm volatile("s_wait_asynccnt 0" ::: "memory");
#endif
}

// Scalar OU coefficients: dz_dt = a*z + b ; dlogp = -D*a (all f32, matches ref algebra)
__device__ __forceinline__ void ou_coeffs(float t, float& a, float& b) {
    float u    = 1.0f - t;
    float e2   = expf(-2.0f * OU_K * u);                       // e^{-2K(1-t)}
    float vart = 1.0f + (OU_SIGMA * OU_SIGMA - 1.0f) * e2;     // 1 - 0.75 e2
    float mt   = expf(-OU_K * u) * OU_M;
    float inv  = 1.0f / vart;
    a = OU_K - OU_K * inv;
    b = OU_K * mt * inv;
}

// ---------------------------------------------------------------------------
// Main streaming kernel: out4[i] = a * z4[i] + b, double-buffered through LDS
// via CDNA5 async global->LDS b128 copies; NT b128 stores (one-touch stream).
// Each wave owns a private LDS slice -> no barriers needed.
// ---------------------------------------------------------------------------
__global__ void __launch_bounds__(256)
ou_flow_kernel(const float* __restrict__ z, const float* __restrict__ t_ptr,
               float* __restrict__ out, long long n4) {
    float a, b;
    ou_coeffs(t_ptr[0], a, b);

    const v4f* z4 = (const v4f*)z;
    v4f*       o4 = (v4f*)out;

    const long long stride = (long long)gridDim.x * (long long)blockDim.x;
    const long long i0     = (long long)blockIdx.x * blockDim.x + threadIdx.x;

#if HAS_ASYNC_LDS
    __shared__ v4f buf[2][256];
    v4f* slot0 = &buf[0][threadIdx.x];
    v4f* slot1 = &buf[1][threadIdx.x];

    if (i0 < n4) {
        // prime the pipeline: tile 0 -> buffer 0
        __builtin_amdgcn_global_load_async_to_lds_b128(GEN_V4I(z4 + i0), AS3_V4I(slot0), 0, 0);
        int cur = 0;
        for (long long i = i0; i < n4; i += stride) {
            const long long nx   = i + stride;
            const bool      more = nx < n4;            // uniform: n4 is a multiple of stride
            if (more) {
                __builtin_amdgcn_global_load_async_to_lds_b128(
                    GEN_V4I(z4 + nx), AS3_V4I(cur ? slot0 : slot1), 0, 0);
                wait_asynccnt_le1();                   // current tile landed in LDS
            } else {
                wait_asynccnt_le0();
            }
            asm volatile("" ::: "memory");             // fence the LDS read below the wait
            v4f v = cur ? *slot1 : *slot0;             // ds_load_b128
            v4f r;
            r.x = fmaf(a, v.x, b);
            r.y = fmaf(a, v.y, b);
            r.z = fmaf(a, v.z, b);
            r.w = fmaf(a, v.w, b);
            __builtin_nontemporal_store(r, &o4[i]);    // global_store_b128 th:NT
            cur ^= 1;
        }
    }
#else
    for (long long i = i0; i < n4; i += stride) {
        v4f v = __builtin_nontemporal_load(&z4[i]);    // global_load_b128 th:NT
        v4f r;
        r.x = fmaf(a, v.x, b);
        r.y = fmaf(a, v.y, b);
        r.z = fmaf(a, v.z, b);
        r.w = fmaf(a, v.w, b);
        __builtin_nontemporal_store(r, &o4[i]);
    }
#endif
}

// ---------------------------------------------------------------------------
// Broadcast kernel: dlogp region = -D * a  (B floats, NT b128 stores)
// ---------------------------------------------------------------------------
__global__ void __launch_bounds__(256)
ou_dlogp_kernel(const float* __restrict__ t_ptr, float* __restrict__ out, long long n4) {
    float a, b;
    ou_coeffs(t_ptr[0], a, b);
    const float val = -(float)OU_D * a;
    v4f r = {val, val, val, val};
    v4f* o4 = (v4f*)out;
    const long long stride = (long long)gridDim.x * (long long)blockDim.x;
    for (long long i = (long long)blockIdx.x * blockDim.x + threadIdx.x; i < n4; i += stride)
        __builtin_nontemporal_store(r, &o4[i]);
}

extern "C" void kernel_launch(void* const* d_in, const int* in_sizes, int n_in,
                              void* d_out, int out_size, void* d_ws, size_t ws_size,
                              hipStream_t stream) {
    (void)n_in; (void)out_size; (void)d_ws; (void)ws_size;
    const float* z = (const float*)d_in[0];
    const float* t = (const float*)d_in[1];
    float*       o = (float*)d_out;

    const long long n     = (long long)in_sizes[0];  // B*D = 67,108,864
    const long long Brows = n / OU_D;                // B   = 131,072
    const long long n4    = n >> 2;                  // 2^24 float4

    const int block = 256;
    long long want  = (n4 + block - 1) / block;
    int grid        = (int)(want < 2048 ? want : 2048);   // 2^24 / (2048*256) = 32 iters exactly
    ou_flow_kernel<<<grid, block, 0, stream>>>(z, t, o, n4);

    const long long nb4 = Brows >> 2;                // 32,768 float4
    long long want2 = (nb4 + block - 1) / block;
    int grid2       = (int)(want2 < 128 ? want2 : 128);
    ou_dlogp_kernel<<<grid2, block, 0, stream>>>(t, o + n, nb4);
}